// RelGat_57836029608136
// MI455X (gfx1250) — compile-verified
//
#include <hip/hip_runtime.h>
#include <hip/hip_bf16.h>
#include <cstddef>

#define DD 128
#define PAD 4
#define NEG_SLOPE_F (((1.0f/8.0f)+(1.0f/3.0f))*0.5f)

typedef float v2f __attribute__((ext_vector_type(2)));
typedef float v8f __attribute__((ext_vector_type(8)));

__device__ __forceinline__ unsigned fkey(float f) {
  unsigned u = __float_as_uint(f);
  return (u & 0x80000000u) ? ~u : (u | 0x80000000u);
}
__device__ __forceinline__ float funkey(unsigned k) {
  unsigned u = (k & 0x80000000u) ? (k & 0x7FFFFFFFu) : ~k;
  return __uint_as_float(u);
}

// ---------------------------------------------------------------------------
// K1: node GEMM. self = h@W2+b2, iso = h@W3+b3.  16 nodes/block, 8 waves,
// wave w -> cols [16w,16w+16). fp32 WMMA 16x16x4, K-loop of 32.
// ---------------------------------------------------------------------------
__global__ __launch_bounds__(256) void node_gemm_kernel(
    const float* __restrict__ h,
    const float* __restrict__ W2, const float* __restrict__ b2,
    const float* __restrict__ W3, const float* __restrict__ b3,
    float* __restrict__ selfm, float* __restrict__ isom)
{
  __shared__ float xs[16][DD + PAD];
  const int tid = threadIdx.x;
  const int n0  = blockIdx.x * 16;
  {
    int r = tid >> 4;
    int cb = (tid & 15) * 8;
    const float* hp = h + (size_t)(n0 + r) * DD + cb;
    float4 a0 = *(const float4*)(hp);
    float4 a1 = *(const float4*)(hp + 4);
    *(float4*)&xs[r][cb]     = a0;
    *(float4*)&xs[r][cb + 4] = a1;
  }
  __syncthreads();

  const int lane = tid & 31;
  const int half = lane >> 4;
  const int l16  = lane & 15;
  const int col0 = (tid >> 5) * 16;

  v8f c2 = {};
  v8f c3 = {};
  const float* W2p = W2 + col0 + l16;
  const float* W3p = W3 + col0 + l16;
#pragma unroll
  for (int kb = 0; kb < 32; ++kb) {
    const int k0 = kb * 4 + 2 * half;
    v2f a; a.x = xs[l16][k0]; a.y = xs[l16][k0 + 1];
    v2f bA; bA.x = W2p[(size_t)k0 * DD]; bA.y = W2p[(size_t)(k0 + 1) * DD];
    v2f bB; bB.x = W3p[(size_t)k0 * DD]; bB.y = W3p[(size_t)(k0 + 1) * DD];
    c2 = __builtin_amdgcn_wmma_f32_16x16x4_f32(false, a, false, bA, (short)0, c2, false, false);
    c3 = __builtin_amdgcn_wmma_f32_16x16x4_f32(false, a, false, bB, (short)0, c3, false, false);
  }
  const float bias2 = b2[col0 + l16];
  const float bias3 = b3[col0 + l16];
#pragma unroll
  for (int j = 0; j < 8; ++j) {
    const int r = j + 8 * half;
    const size_t idx = (size_t)(n0 + r) * DD + col0 + l16;
    selfm[idx] = c2[j] + bias2;
    isom[idx]  = c3[j] + bias3;
  }
}

// ---------------------------------------------------------------------------
// K2: edge GEMM. m = (h[src]+efeat)@W1+b1 ; score = <h[dst], m> (LDS reduce).
// 64 edges/block, 8 waves. Each wave holds its full K=128 B-slice of W1 in
// 64 VGPRs (v2f bf[32]) and reuses it across 4 row tiles -> 128 WMMA/wave.
// ---------------------------------------------------------------------------
__global__ __launch_bounds__(256) void edge_gemm_kernel(
    const float* __restrict__ h, const float* __restrict__ ef,
    const int* __restrict__ src, const int* __restrict__ dst,
    const float* __restrict__ W1, const float* __restrict__ b1,
    float* __restrict__ m, float* __restrict__ score)
{
  __shared__ float xs[64][DD + PAD];   // h[src]+efeat
  __shared__ float sc[64];
  const int tid = threadIdx.x;
  const int e0  = blockIdx.x * 64;

  // stage A: 64 gathered rows of h[src]+efeat
#pragma unroll
  for (int rr = 0; rr < 4; ++rr) {
    int r = rr * 16 + (tid >> 4);
    int cb = (tid & 15) * 8;
    int e = e0 + r;
    const float* hs = h + (size_t)src[e] * DD + cb;
    const float* ep = ef + (size_t)e * DD + cb;
    float4 a0 = *(const float4*)(hs);
    float4 a1 = *(const float4*)(hs + 4);
    float4 f0 = *(const float4*)(ep);
    float4 f1 = *(const float4*)(ep + 4);
    float4 x0, x1;
    x0.x = a0.x + f0.x; x0.y = a0.y + f0.y; x0.z = a0.z + f0.z; x0.w = a0.w + f0.w;
    x1.x = a1.x + f1.x; x1.y = a1.y + f1.y; x1.z = a1.z + f1.z; x1.w = a1.w + f1.w;
    *(float4*)&xs[r][cb]     = x0;
    *(float4*)&xs[r][cb + 4] = x1;
  }
  if (tid < 64) sc[tid] = 0.0f;
  __syncthreads();

  const int lane = tid & 31;
  const int half = lane >> 4;
  const int l16  = lane & 15;
  const int col0 = (tid >> 5) * 16;

  // load the wave's 16-column B slice of W1 once (held in registers)
  v2f bf[32];
  {
    const float* Wp = W1 + col0 + l16;
#pragma unroll
    for (int kb = 0; kb < 32; ++kb) {
      const int k0 = kb * 4 + 2 * half;
      bf[kb].x = Wp[(size_t)k0 * DD];
      bf[kb].y = Wp[(size_t)(k0 + 1) * DD];
    }
  }
  const float bias = b1[col0 + l16];

#pragma unroll
  for (int rt = 0; rt < 4; ++rt) {
    const int rbase = rt * 16;
    v8f c = {};
#pragma unroll
    for (int kb = 0; kb < 32; ++kb) {
      const int k0 = kb * 4 + 2 * half;
      v2f a; a.x = xs[rbase + l16][k0]; a.y = xs[rbase + l16][k0 + 1];
      c = __builtin_amdgcn_wmma_f32_16x16x4_f32(false, a, false, bf[kb], (short)0, c, false, false);
    }
#pragma unroll
    for (int j = 0; j < 8; ++j) {
      const int r = rbase + j + 8 * half;
      const float val = c[j] + bias;
      m[(size_t)(e0 + r) * DD + col0 + l16] = val;
      const int dn = dst[e0 + r];
      atomicAdd(&sc[r], val * h[(size_t)dn * DD + col0 + l16]);
    }
  }
  __syncthreads();
  if (tid < 64) score[e0 + tid] = sc[tid];
}

// ---------------------------------------------------------------------------
// K3: zero a float range (used for smaxkey / denom / wsum, contiguous in ws)
// ---------------------------------------------------------------------------
__global__ void zero_kernel(float* __restrict__ p, long long n)
{
  long long i = (long long)blockIdx.x * blockDim.x + threadIdx.x;
  if (i < n) p[i] = 0.0f;
}

// K4: segment-max via monotonic uint atomicMax
__global__ void smax_kernel(const float* __restrict__ score, const int* __restrict__ dst,
                            unsigned* __restrict__ key, int E)
{
  int e = blockIdx.x * blockDim.x + threadIdx.x;
  if (e >= E) return;
  atomicMax(&key[dst[e]], fkey(score[e]));
}

// K5: w = exp(score - smax[dst]); denom[dst] += w
__global__ void weight_kernel(const float* __restrict__ score, const int* __restrict__ dst,
                              const unsigned* __restrict__ key, float* __restrict__ w,
                              float* __restrict__ denom, int E)
{
  int e = blockIdx.x * blockDim.x + threadIdx.x;
  if (e >= E) return;
  int dn = dst[e];
  float wv = expf(score[e] - funkey(key[dn]));
  w[e] = wv;
  atomicAdd(&denom[dn], wv);
}

// K6: wsum[dst] += w * m   (one wave per edge, float4 per lane)
__global__ __launch_bounds__(256) void scatter_kernel(
    const float* __restrict__ m, const float* __restrict__ w,
    const int* __restrict__ dst, float* __restrict__ wsum, int E)
{
  int t = blockIdx.x * blockDim.x + threadIdx.x;
  int e = t >> 5;
  int lane = t & 31;
  if (e >= E) return;
  const float wv = w[e];
  const int dn = dst[e];
  const float4 mv = *(const float4*)(m + (size_t)e * DD + lane * 4);
  float* out = wsum + (size_t)dn * DD + lane * 4;
  atomicAdd(out + 0, wv * mv.x);
  atomicAdd(out + 1, wv * mv.y);
  atomicAdd(out + 2, wv * mv.z);
  atomicAdd(out + 3, wv * mv.w);
}

// K7: combine + leaky (RReLU eval).  denom==0 <=> deg==0 (w>0 always).
__global__ __launch_bounds__(256) void combine_kernel(
    const float* __restrict__ selfm, const float* __restrict__ isom,
    const float* __restrict__ wsum, const float* __restrict__ denom,
    float* __restrict__ hout, int N)
{
  int t = blockIdx.x * blockDim.x + threadIdx.x;
  int n = t >> 5;
  int lane = t & 31;
  if (n >= N) return;
  const float d = denom[n];
  const float inv = 1.0f / fmaxf(d, 1e-30f);
  const size_t base = (size_t)n * DD + lane * 4;
  const float* sel = (d == 0.0f) ? isom : selfm;
  float4 s = *(const float4*)(sel + base);
  float4 a = *(const float4*)(wsum + base);
  float4 x;
  x.x = s.x + a.x * inv; x.y = s.y + a.y * inv;
  x.z = s.z + a.z * inv; x.w = s.w + a.w * inv;
  x.x = (x.x >= 0.0f) ? x.x : x.x * NEG_SLOPE_F;
  x.y = (x.y >= 0.0f) ? x.y : x.y * NEG_SLOPE_F;
  x.z = (x.z >= 0.0f) ? x.z : x.z * NEG_SLOPE_F;
  x.w = (x.w >= 0.0f) ? x.w : x.w * NEG_SLOPE_F;
  *(float4*)(hout + base) = x;
}

// ---------------------------------------------------------------------------
extern "C" void kernel_launch(void* const* d_in, const int* in_sizes, int n_in,
                              void* d_out, int out_size, void* d_ws, size_t ws_size,
                              hipStream_t stream)
{
  const float* node_feats = (const float*)d_in[0];
  const float* edge_feats = (const float*)d_in[1];
  const int*   src        = (const int*)d_in[2];
  const int*   dst        = (const int*)d_in[3];
  const int E = in_sizes[2];
  const int N = in_sizes[0] / DD;

  const float* W[2][3];
  const float* B[2][3];
  for (int l = 0; l < 2; ++l)
    for (int w = 0; w < 3; ++w) {
      W[l][w] = (const float*)d_in[4 + l * 6 + w * 2];
      B[l][w] = (const float*)d_in[4 + l * 6 + w * 2 + 1];
    }

  // workspace layout (floats)
  float* wsf = (float*)d_ws;
  size_t off = 0;
  float*    m_buf   = wsf + off; off += (size_t)E * DD;
  float*    score   = wsf + off; off += (size_t)E;
  float*    w_buf   = wsf + off; off += (size_t)E;
  unsigned* smaxkey = (unsigned*)(wsf + off); off += (size_t)N;
  float*    denom   = wsf + off; off += (size_t)N;
  float*    wsum    = wsf + off; off += (size_t)N * DD;
  float*    selfm   = wsf + off; off += (size_t)N * DD;
  float*    isom    = wsf + off; off += (size_t)N * DD;
  float*    h1      = wsf + off; off += (size_t)N * DD;

  const long long zcount = (long long)N * (DD + 2); // smaxkey + denom + wsum contiguous
  const int zblocks = (int)((zcount + 255) / 256);

  auto layer = [&](const float* h, const float* W1, const float* b1,
                   const float* W2, const float* b2,
                   const float* W3, const float* b3, float* hout) {
    node_gemm_kernel<<<N / 16, 256, 0, stream>>>(h, W2, b2, W3, b3, selfm, isom);
    zero_kernel<<<zblocks, 256, 0, stream>>>((float*)smaxkey, zcount);
    edge_gemm_kernel<<<E / 64, 256, 0, stream>>>(h, edge_feats, src, dst, W1, b1, m_buf, score);
    smax_kernel<<<(E + 255) / 256, 256, 0, stream>>>(score, dst, smaxkey, E);
    weight_kernel<<<(E + 255) / 256, 256, 0, stream>>>(score, dst, smaxkey, w_buf, denom, E);
    scatter_kernel<<<(E * 32 + 255) / 256, 256, 0, stream>>>(m_buf, w_buf, dst, wsum, E);
    combine_kernel<<<(N * 32 + 255) / 256, 256, 0, stream>>>(selfm, isom, wsum, denom, hout, N);
  };

  layer(node_feats, W[0][0], B[0][0], W[0][1], B[0][1], W[0][2], B[0][2], h1);
  layer(h1,         W[1][0], B[1][0], W[1][1], B[1][1], W[1][2], B[1][2], (float*)d_out);
}